// TriangleDistance_42872363549145
// MI455X (gfx1250) — compile-verified
//
#include <hip/hip_runtime.h>

typedef float v2f __attribute__((ext_vector_type(2)));
typedef float v8f __attribute__((ext_vector_type(8)));

#define TRI_EPS 1e-12f

// Per-(triangle, lane-half) record, 96 bytes, 16B aligned.
// A wave32 lane in the low half (lanes 0-15) consumes K0,K1 components of the
// six 4x16 B matrices; a high-half lane consumes K2,K3. We pre-split so each
// lane does straight b128 loads with no cndmask selects.
struct __align__(16) TriHalf {
    float4 m01;  // (Bab.k0k1 | Bab.k2k3 , Bac.k0k1 | Bac.k2k3)
    float4 m23;  // (Bn  ...            , Ba  ...             )
    float4 m45;  // (Bb  ...            , Bc  ...             )
    float4 k0;   // (a.a, b.b, c.c, |ab|^2)             [same both halves]
    float4 k1;   // (ab.ac, |ac|^2, 1/max(|ab|^2,eps), 1/max(|ac|^2,eps))
    float2 k2;   // (1/max(|bc|^2,eps), 1/max(|n|^2,eps))
    float2 pad;
};

__global__ void tri_prep_kernel(const float* __restrict__ va,
                                const float* __restrict__ vb,
                                const float* __restrict__ vc,
                                TriHalf* __restrict__ pre, int M) {
    int j = blockIdx.x * blockDim.x + threadIdx.x;
    if (j >= M) return;
    float ax = va[3*j], ay = va[3*j+1], az = va[3*j+2];
    float bx = vb[3*j], by = vb[3*j+1], bz = vb[3*j+2];
    float cx = vc[3*j], cy = vc[3*j+1], cz = vc[3*j+2];
    float abx = bx-ax, aby = by-ay, abz = bz-az;
    float acx = cx-ax, acy = cy-ay, acz = cz-az;
    float bcx = cx-bx, bcy = cy-by, bcz = cz-bz;
    float nx = aby*acz - abz*acy;
    float ny = abz*acx - abx*acz;
    float nz = abx*acy - aby*acx;
    float a_ab = ax*abx + ay*aby + az*abz;
    float a_ac = ax*acx + ay*acy + az*acz;
    float a_n  = ax*nx  + ay*ny  + az*nz;
    float a2 = ax*ax + ay*ay + az*az;
    float b2 = bx*bx + by*by + bz*bz;
    float c2 = cx*cx + cy*cy + cz*cz;
    float ab2  = abx*abx + aby*aby + abz*abz;
    float ac2  = acx*acx + acy*acy + acz*acz;
    float abac = abx*acx + aby*acy + abz*acz;
    float bc2  = bcx*bcx + bcy*bcy + bcz*bcz;
    float nn   = nx*nx + ny*ny + nz*nz;

    const float4 k0 = make_float4(a2, b2, c2, ab2);
    const float4 k1 = make_float4(abac, ac2, 1.0f/fmaxf(ab2, TRI_EPS), 1.0f/fmaxf(ac2, TRI_EPS));
    const float2 k2 = make_float2(1.0f/fmaxf(bc2, TRI_EPS), 1.0f/fmaxf(nn, TRI_EPS));

    // B columns (4-vectors): Bab=(ab,-a.ab) Bac=(ac,-a.ac) Bn=(n,-a.n)
    //                        Ba=(-2a,1) Bb=(-2b,1) Bc=(-2c,1)
    TriHalf lo, hi;
    lo.m01 = make_float4(abx, aby,           acx, acy);
    lo.m23 = make_float4(nx,  ny,      -2.0f*ax, -2.0f*ay);
    lo.m45 = make_float4(-2.0f*bx, -2.0f*by, -2.0f*cx, -2.0f*cy);
    hi.m01 = make_float4(abz, -a_ab,         acz, -a_ac);
    hi.m23 = make_float4(nz,  -a_n,    -2.0f*az, 1.0f);
    hi.m45 = make_float4(-2.0f*bz, 1.0f,     -2.0f*cz, 1.0f);
    lo.k0 = hi.k0 = k0;
    lo.k1 = hi.k1 = k1;
    lo.k2 = hi.k2 = k2;
    lo.pad = hi.pad = make_float2(0.0f, 0.0f);
    pre[2*j + 0] = lo;
    pre[2*j + 1] = hi;
}

static __device__ __forceinline__ v8f wmma4(v2f a, v2f b, v8f c) {
    // V_WMMA_F32_16X16X4_F32: D = A(16x4) * B(4x16) + C(16x16), wave32.
    return __builtin_amdgcn_wmma_f32_16x16x4_f32(
        /*neg_a=*/false, a, /*neg_b=*/false, b,
        /*c_mod=*/(short)0, c, /*reuse_a=*/false, /*reuse_b=*/false);
}

static __device__ __forceinline__ v2f xy(const float4 f) { v2f r; r.x = f.x; r.y = f.y; return r; }
static __device__ __forceinline__ v2f zw(const float4 f) { v2f r; r.x = f.z; r.y = f.w; return r; }

__global__ void __launch_bounds__(256)
tri_dist_kernel(const float* __restrict__ pts,
                const TriHalf* __restrict__ pre,
                float* __restrict__ dminO,
                long long* __restrict__ idxO,
                long long* __restrict__ typO,
                int N, int M) {
    const int lane   = threadIdx.x & 31;
    const int wave   = blockIdx.x * (blockDim.x >> 5) + (threadIdx.x >> 5);
    const int ptBase = wave * 16;                  // 16-point tile per wave
    if (ptBase >= N) return;                       // wave-uniform exit
    const int h   = (lane >> 4) & 1;               // 0: K0K1 half, 1: K2K3 half
    const int col = lane & 15;                     // triangle column within tile

    // A operands: lane holds point m = ptBase + (lane&15).
    // Low half supplies K0,K1=(x,y); high half K2,K3=(z,w).
    const int m = ptBase + col;
    const float px = pts[3*m + 0];
    const float py = pts[3*m + 1];
    const float pz = pts[3*m + 2];
    const float p2 = px*px + py*py + pz*pz;
    v2f A1, A2;                                    // A1: w=1 ; A2: w=|p|^2
    A1.x = h ? pz : px;  A1.y = h ? 1.0f : py;
    A2.x = h ? pz : px;  A2.y = h ? p2   : py;

    float best[8]; int bpk[8];                     // bpk = (tri_idx<<3) | type
#pragma unroll
    for (int r = 0; r < 8; ++r) { best[r] = 3.402823466e38f; bpk[r] = 0; }

    const v8f zero = {0,0,0,0,0,0,0,0};
    const int laneOff   = 2*col + h;               // record index within a tile row
    const int numTiles  = M >> 4;
    const TriHalf* __restrict__ Tp = pre + laneOff;

    for (int t = 0; t < numTiles; ++t, Tp += 32) {
        const int j = (t << 4) + col;              // this lane's triangle column
        const float4 m01 = Tp->m01;                // 3 x b128: six v2f B-halves
        const float4 m23 = Tp->m23;
        const float4 m45 = Tp->m45;
        const float4 k0  = Tp->k0;
        const float4 k1  = Tp->k1;
        const float2 k2  = Tp->k2;

        // C operands broadcast per column (same value in all 8 row-VGPRs).
        const v8f Ca = {k0.x,k0.x,k0.x,k0.x,k0.x,k0.x,k0.x,k0.x};
        const v8f Cb = {k0.y,k0.y,k0.y,k0.y,k0.y,k0.y,k0.y,k0.y};
        const v8f Cc = {k0.z,k0.z,k0.z,k0.z,k0.z,k0.z,k0.z,k0.z};

        const v8f d1  = wmma4(A1, xy(m01), zero);  // ab.(p-a)
        const v8f d2  = wmma4(A1, zw(m01), zero);  // ac.(p-a)
        const v8f apn = wmma4(A1, xy(m23), zero);  // n.(p-a)
        const v8f dpa = wmma4(A2, zw(m23), Ca);    // |p-a|^2
        const v8f dpb = wmma4(A2, xy(m45), Cb);    // |p-b|^2
        const v8f dpc = wmma4(A2, zw(m45), Cc);    // |p-c|^2

        const float ab2 = k0.w, abac = k1.x, ac2 = k1.y;
        const float inv_ab2 = k1.z, inv_ac2 = k1.w;
        const float inv_bc2 = k2.x, inv_nn = k2.y;

#pragma unroll
        for (int r = 0; r < 8; ++r) {
            const float d1e = d1[r], d2e = d2[r], ape = apn[r];
            const float da = dpa[r], db = dpb[r], dc = dpc[r];
            const float d3e = d1e - ab2;
            const float d4e = d2e - abac;
            const float d5e = d1e - abac;
            const float d6e = d2e - ac2;
            const float va = d3e*d6e - d5e*d4e;
            const float vb = d5e*d2e - d1e*d6e;
            const float vc = d1e*d4e - d3e*d2e;
            const float nbc = d4e - d3e;
            const float d56 = d5e - d6e;

            float dist = ape * ape * inv_nn;       // face interior
            int   typ  = 6;
            if (va <= 0.0f && nbc >= 0.0f && d56 >= 0.0f) { dist = db - nbc*nbc*inv_bc2; typ = 4; }
            if (vb <= 0.0f && d2e >= 0.0f && d6e <= 0.0f) { dist = da - d2e*d2e*inv_ac2; typ = 5; }
            if (d6e >= 0.0f && d5e <= d6e)                { dist = dc;                   typ = 2; }
            if (vc <= 0.0f && d1e >= 0.0f && d3e <= 0.0f) { dist = da - d1e*d1e*inv_ab2; typ = 3; }
            if (d3e >= 0.0f && d4e <= d3e)                { dist = db;                   typ = 1; }
            if (d1e <= 0.0f && d2e <= 0.0f)               { dist = da;                   typ = 0; }
            dist = fmaxf(dist, 0.0f);

            if (dist < best[r]) { best[r] = dist; bpk[r] = (j << 3) | typ; }
        }
    }

    // Reduce across the 16 columns held by each 16-lane half (rows live in
    // fixed VGPR slots). Packed (idx<<3)|typ preserves first-index tie-break.
#pragma unroll
    for (int r = 0; r < 8; ++r) {
        float d = best[r]; int pk = bpk[r];
        for (int off = 1; off < 16; off <<= 1) {   // stays within 16-lane halves
            const float od = __shfl_xor(d, off, 32);
            const int   op = __shfl_xor(pk, off, 32);
            if (od < d || (od == d && op < pk)) { d = od; pk = op; }
        }
        if (col == 0) {                            // lane 0 -> rows 0..7, lane 16 -> rows 8..15
            const int p = ptBase + r + (h ? 8 : 0);
            dminO[p] = d;
            idxO[p]  = (long long)(pk >> 3);
            typO[p]  = (long long)(pk & 7);
        }
    }
}

extern "C" void kernel_launch(void* const* d_in, const int* in_sizes, int n_in,
                              void* d_out, int out_size, void* d_ws, size_t ws_size,
                              hipStream_t stream) {
    const float* pts = (const float*)d_in[0];
    const float* v1  = (const float*)d_in[1];
    const float* v2  = (const float*)d_in[2];
    const float* v3  = (const float*)d_in[3];
    const int N = in_sizes[0] / 3;   // 8192 points
    const int M = in_sizes[1] / 3;   // 2048 triangles (multiple of 16)

    TriHalf* pre = (TriHalf*)d_ws;   // 96 B * 2 * M = 384 KB scratch

    float*     dminO = (float*)d_out;
    long long* idxO  = (long long*)(dminO + N);
    long long* typO  = idxO + N;

    tri_prep_kernel<<<(M + 255) / 256, 256, 0, stream>>>(v1, v2, v3, pre, M);

    const int waves   = N / 16;                  // one wave32 per 16-point tile
    const int threads = 256;                     // 8 waves per block
    const int blocks  = (waves * 32 + threads - 1) / threads;
    tri_dist_kernel<<<blocks, threads, 0, stream>>>(pts, pre, dminO, idxO, typO, N, M);
}